// GraphEncoder_18382460027182
// MI455X (gfx1250) — compile-verified
//
#include <hip/hip_runtime.h>
#include <hip/hip_bf16.h>

// ---------------------------------------------------------------------------
// GIN graph encoder for MI455X (gfx1250, wave32).
//   agg = segment_sum(x[src], dst)                      (L2-resident scatter)
//   h   = relu(((1+eps)x + agg) @ W1 + b1) @ W2 + b2    (fp32 WMMA 16x16x4,
//                                                        A in LDS, 32-row tiles)
//   h   = FiLM(softplus) -> SiLU -> mean pool by graph  (fused into GEMM2)
// ---------------------------------------------------------------------------

#define N_NODES 50000
#define E_EDGES 1600000
#define IN_DIM  128
#define D_H     256
#define N_GRAPHS 512

// LDS row pitch padding: bank = ((pitch*row + k) % 64); pitch%64 == 4 spreads
// the 16 row-lanes of a half-wave across distinct banks.
#define PITCH1 (IN_DIM + 4)   // 132, multiple of 4 -> float4-aligned rows
#define PITCH2 (D_H + 4)      // 260, multiple of 4 -> float4-aligned rows

typedef __attribute__((ext_vector_type(2))) float v2f;
typedef __attribute__((ext_vector_type(8))) float v8f;

// -------------------------- edge scatter-add -------------------------------
// One wave (32 lanes) per edge; lane l moves features [4l, 4l+4).
// x rows are 512B -> the float4 gather is a coalesced 512B burst per wave;
// adds are hardware global_atomic_add_f32 hitting the 192MB L2.
__global__ __launch_bounds__(256) void scatter_add_kernel(
    const float* __restrict__ x, const int* __restrict__ edge_index,
    float* __restrict__ agg) {
  const int tid  = blockIdx.x * blockDim.x + threadIdx.x;
  const int edge = tid >> 5;
  const int lane = tid & 31;
  if (edge >= E_EDGES) return;
  const int s = edge_index[edge];            // edge_index[0] row = src
  const int d = edge_index[E_EDGES + edge];  // edge_index[1] row = dst
  const float4 v = *(const float4*)(x + (size_t)s * IN_DIM + lane * 4);
  float* a = agg + (size_t)d * IN_DIM + lane * 4;
  unsafeAtomicAdd(a + 0, v.x);
  unsafeAtomicAdd(a + 1, v.y);
  unsafeAtomicAdd(a + 2, v.z);
  unsafeAtomicAdd(a + 3, v.w);
}

// ------------------------------ GEMM 1 -------------------------------------
// h1 = relu(((1+eps)*x + agg) @ W1 + b1).  One block = one 32-row M-tile over
// all 256 output columns.  The 32x128 combined activation tile is computed
// once into LDS; each of 8 waves produces a 32x32 output patch as four 16x16
// WMMA tiles (2 M-subtiles x 2 N-tiles), so every B fragment feeds two
// V_WMMA_F32_16X16X4_F32 ops.  A-fragment layout (16x4 f32): lane<16 holds
// (M=lane, K={0,1}), lane>=16 holds (M=lane-16, K={2,3}); B mirrors with N.
__global__ __launch_bounds__(256) void gemm1_kernel(
    const float* __restrict__ x, const float* __restrict__ agg,
    const float* __restrict__ W1, const float* __restrict__ b1,
    const float* __restrict__ eps_p, float* __restrict__ h1) {
  __shared__ float As[32 * PITCH1];

  const int rowBase = blockIdx.x * 32;
  const float eps1 = 1.0f + eps_p[0];

  // Stage (1+eps)*x + agg for 32 rows into LDS (1024 float4 / 256 threads).
  // Source row clamped for the ragged last block (50000 = 32*1562 + 16).
#pragma unroll
  for (int it = 0; it < 4; ++it) {
    const int i = it * 256 + threadIdx.x;
    const int row = i >> 5;            // 32 float4 per 128-float row
    const int q   = (i & 31) * 4;
    int srcRow = rowBase + row;
    srcRow = srcRow < N_NODES ? srcRow : N_NODES - 1;
    const float4 xv = *(const float4*)(x   + (size_t)srcRow * IN_DIM + q);
    const float4 av = *(const float4*)(agg + (size_t)srcRow * IN_DIM + q);
    float4 m;
    m.x = eps1 * xv.x + av.x;  m.y = eps1 * xv.y + av.y;
    m.z = eps1 * xv.z + av.z;  m.w = eps1 * xv.w + av.w;
    *(float4*)(As + row * PITCH1 + q) = m;
  }
  __syncthreads();

  const int lane = threadIdx.x & 31;
  const int wave = threadIdx.x >> 5;
  const int half = lane >> 4;          // 0 or 1
  const int koff = half * 2;           // K sub-offset {0,2}
  const int mr   = lane & 15;          // A row / B col within tile
  const int n0   = wave * 32 + mr;     // two N-tiles per wave: n0, n0+16
  const int n1   = n0 + 16;

  v8f c00, c01, c10, c11;
  const float bias0 = b1[n0], bias1 = b1[n1];
#pragma unroll
  for (int r = 0; r < 8; ++r) {
    c00[r] = bias0; c01[r] = bias1; c10[r] = bias0; c11[r] = bias1;
  }

  const float* as0 = As + mr * PITCH1 + koff;            // M-subtile 0
  const float* as1 = as0 + 16 * PITCH1;                  // M-subtile 1
#pragma unroll 2
  for (int k0 = 0; k0 < IN_DIM; k0 += 4) {
    const int k = k0 + koff;
    const v2f a0 = *(const v2f*)(as0 + k0);              // LDS, conflict-free
    const v2f a1 = *(const v2f*)(as1 + k0);
    v2f b0; b0.x = W1[(size_t)k * D_H + n0]; b0.y = W1[(size_t)(k + 1) * D_H + n0];
    v2f b1v; b1v.x = W1[(size_t)k * D_H + n1]; b1v.y = W1[(size_t)(k + 1) * D_H + n1];
    c00 = __builtin_amdgcn_wmma_f32_16x16x4_f32(false, a0, false, b0,
                                                (short)0, c00, false, false);
    c01 = __builtin_amdgcn_wmma_f32_16x16x4_f32(false, a0, false, b1v,
                                                (short)0, c01, false, false);
    c10 = __builtin_amdgcn_wmma_f32_16x16x4_f32(false, a1, false, b0,
                                                (short)0, c10, false, false);
    c11 = __builtin_amdgcn_wmma_f32_16x16x4_f32(false, a1, false, b1v,
                                                (short)0, c11, false, false);
  }

  // C/D layout: VGPR r -> lanes 0-15 (M=r, N=lane), lanes 16-31 (M=r+8).
#pragma unroll
  for (int r = 0; r < 8; ++r) {
    const int m = r + half * 8;
    float* hrow0 = h1 + (size_t)(rowBase + m) * D_H;
    hrow0[n0] = c00[r] > 0.0f ? c00[r] : 0.0f;           // ReLU
    hrow0[n1] = c01[r] > 0.0f ? c01[r] : 0.0f;
    if (rowBase + 16 + m < N_NODES) {                    // ragged last block
      float* hrow1 = h1 + (size_t)(rowBase + 16 + m) * D_H;
      hrow1[n0] = c10[r] > 0.0f ? c10[r] : 0.0f;
      hrow1[n1] = c11[r] > 0.0f ? c11[r] : 0.0f;
    }
  }
}

// ------------------------------ GEMM 2 + FiLM + SiLU + pool ----------------
// t = h1 @ W2 + b2 ; h = (1+softplus(gamma)*ng)*t + softplus(beta)*nb ;
// h = h*sigmoid(h) ; sums[batch[row]] += h   (fused, no h2 materialized).
__global__ __launch_bounds__(256) void gemm2_kernel(
    const float* __restrict__ h1, const float* __restrict__ W2,
    const float* __restrict__ b2, const float* __restrict__ gamma,
    const float* __restrict__ beta, const float* __restrict__ noise_g,
    const float* __restrict__ noise_b, const int* __restrict__ batch,
    float* __restrict__ sums) {
  __shared__ float As[32 * PITCH2];

  const int rowBase = blockIdx.x * 32;

  // Stage 32x256 h1 tile into LDS (2048 float4 / 256 threads).
#pragma unroll
  for (int it = 0; it < 8; ++it) {
    const int i = it * 256 + threadIdx.x;
    const int row = i >> 6;            // 64 float4 per 256-float row
    const int q   = (i & 63) * 4;
    int srcRow = rowBase + row;
    srcRow = srcRow < N_NODES ? srcRow : N_NODES - 1;
    *(float4*)(As + row * PITCH2 + q) =
        *(const float4*)(h1 + (size_t)srcRow * D_H + q);
  }
  __syncthreads();

  const int lane = threadIdx.x & 31;
  const int wave = threadIdx.x >> 5;
  const int half = lane >> 4;
  const int koff = half * 2;
  const int mr   = lane & 15;
  const int n0   = wave * 32 + mr;
  const int n1   = n0 + 16;

  v8f c00, c01, c10, c11;
  const float bias0 = b2[n0], bias1 = b2[n1];
#pragma unroll
  for (int r = 0; r < 8; ++r) {
    c00[r] = bias0; c01[r] = bias1; c10[r] = bias0; c11[r] = bias1;
  }

  const float* as0 = As + mr * PITCH2 + koff;
  const float* as1 = as0 + 16 * PITCH2;
#pragma unroll 2
  for (int k0 = 0; k0 < D_H; k0 += 4) {
    const int k = k0 + koff;
    const v2f a0 = *(const v2f*)(as0 + k0);
    const v2f a1 = *(const v2f*)(as1 + k0);
    v2f b0; b0.x = W2[(size_t)k * D_H + n0]; b0.y = W2[(size_t)(k + 1) * D_H + n0];
    v2f b1v; b1v.x = W2[(size_t)k * D_H + n1]; b1v.y = W2[(size_t)(k + 1) * D_H + n1];
    c00 = __builtin_amdgcn_wmma_f32_16x16x4_f32(false, a0, false, b0,
                                                (short)0, c00, false, false);
    c01 = __builtin_amdgcn_wmma_f32_16x16x4_f32(false, a0, false, b1v,
                                                (short)0, c01, false, false);
    c10 = __builtin_amdgcn_wmma_f32_16x16x4_f32(false, a1, false, b0,
                                                (short)0, c10, false, false);
    c11 = __builtin_amdgcn_wmma_f32_16x16x4_f32(false, a1, false, b1v,
                                                (short)0, c11, false, false);
  }

  // Per-feature FiLM params (depend only on the output column).
  const float gs0 = 1.0f + log1pf(__expf(gamma[n0])) * noise_g[n0];
  const float bs0 = log1pf(__expf(beta[n0])) * noise_b[n0];
  const float gs1 = 1.0f + log1pf(__expf(gamma[n1])) * noise_g[n1];
  const float bs1 = log1pf(__expf(beta[n1])) * noise_b[n1];

#pragma unroll
  for (int r = 0; r < 8; ++r) {
    const int m = r + half * 8;
    {
      const int gid = batch[rowBase + m];
      float t0 = gs0 * c00[r] + bs0;
      t0 = t0 * (1.0f / (1.0f + __expf(-t0)));           // SiLU
      float t1 = gs1 * c01[r] + bs1;
      t1 = t1 * (1.0f / (1.0f + __expf(-t1)));
      unsafeAtomicAdd(&sums[(size_t)gid * D_H + n0], t0);
      unsafeAtomicAdd(&sums[(size_t)gid * D_H + n1], t1);
    }
    if (rowBase + 16 + m < N_NODES) {                    // ragged last block
      const int gid = batch[rowBase + 16 + m];
      float t0 = gs0 * c10[r] + bs0;
      t0 = t0 * (1.0f / (1.0f + __expf(-t0)));
      float t1 = gs1 * c11[r] + bs1;
      t1 = t1 * (1.0f / (1.0f + __expf(-t1)));
      unsafeAtomicAdd(&sums[(size_t)gid * D_H + n0], t0);
      unsafeAtomicAdd(&sums[(size_t)gid * D_H + n1], t1);
    }
  }
}

// ------------------------------ pooling tail -------------------------------
__global__ __launch_bounds__(256) void counts_kernel(
    const int* __restrict__ batch, float* __restrict__ counts) {
  const int i = blockIdx.x * blockDim.x + threadIdx.x;
  if (i < N_NODES) unsafeAtomicAdd(&counts[batch[i]], 1.0f);
}

__global__ __launch_bounds__(256) void finalize_kernel(
    const float* __restrict__ sums, const float* __restrict__ counts,
    float* __restrict__ out) {
  const int i = blockIdx.x * blockDim.x + threadIdx.x;
  if (i >= N_GRAPHS * D_H) return;
  const float cnt = counts[i >> 8];                  // i / D_H
  out[i] = sums[i] / (cnt > 1.0f ? cnt : 1.0f);
}

// ---------------------------------------------------------------------------
extern "C" void kernel_launch(void* const* d_in, const int* in_sizes, int n_in,
                              void* d_out, int out_size, void* d_ws, size_t ws_size,
                              hipStream_t stream) {
  const float* x       = (const float*)d_in[0];
  const int*   eidx    = (const int*)d_in[1];
  const int*   batch   = (const int*)d_in[2];
  const float* W1      = (const float*)d_in[3];
  const float* b1      = (const float*)d_in[4];
  const float* W2      = (const float*)d_in[5];
  const float* b2      = (const float*)d_in[6];
  const float* eps_gin = (const float*)d_in[7];
  const float* gamma   = (const float*)d_in[8];
  const float* beta    = (const float*)d_in[9];
  const float* noise_g = (const float*)d_in[10];
  const float* noise_b = (const float*)d_in[11];
  float* out = (float*)d_out;

  // Workspace layout (fp32): agg | h1 | sums | counts  (~77.3 MB)
  float* agg    = (float*)d_ws;                         // N*128
  float* h1     = agg + (size_t)N_NODES * IN_DIM;       // N*256
  float* sums   = h1 + (size_t)N_NODES * D_H;           // G*256
  float* counts = sums + (size_t)N_GRAPHS * D_H;        // G

  hipMemsetAsync(agg, 0, (size_t)N_NODES * IN_DIM * sizeof(float), stream);
  hipMemsetAsync(sums, 0, ((size_t)N_GRAPHS * D_H + N_GRAPHS) * sizeof(float),
                 stream);

  // 1) edge scatter-add: one wave per edge -> E*32 threads
  {
    const long long threads = (long long)E_EDGES * 32;
    const int blocks = (int)((threads + 255) / 256);
    scatter_add_kernel<<<blocks, 256, 0, stream>>>(x, eidx, agg);
  }

  const int mBlocks = (N_NODES + 31) / 32;              // 1563 (last ragged)

  // 2) GEMM1: one block per 32-row M-tile, A staged in LDS
  gemm1_kernel<<<mBlocks, 256, 0, stream>>>(x, agg, W1, b1, eps_gin, h1);

  // 3) counts (independent of 2; same stream keeps required ordering)
  counts_kernel<<<(N_NODES + 255) / 256, 256, 0, stream>>>(batch, counts);

  // 4) GEMM2 fused FiLM + SiLU + atomic graph pooling
  gemm2_kernel<<<mBlocks, 256, 0, stream>>>(h1, W2, b2, gamma, beta,
                                            noise_g, noise_b, batch, sums);

  // 5) mean divide
  finalize_kernel<<<(N_GRAPHS * D_H + 255) / 256, 256, 0, stream>>>(sums,
                                                                    counts, out);
}